// GraphResidualClassifier_30657476559141
// MI455X (gfx1250) — compile-verified
//
#include <hip/hip_runtime.h>
#include <math.h>

typedef __attribute__((ext_vector_type(2))) float v2f;
typedef __attribute__((ext_vector_type(8))) float v8f;

#define D 128
#define LEAKY 0.2f

// ---- order-preserving float<->uint transform for exact atomic segment-max ----
__device__ __forceinline__ unsigned f2ord(float f) {
    unsigned u = __float_as_uint(f);
    return (u & 0x80000000u) ? ~u : (u | 0x80000000u);
}
__device__ __forceinline__ float ord2f(unsigned u) {
    unsigned b = (u & 0x80000000u) ? (u & 0x7FFFFFFFu) : ~u;
    return __uint_as_float(b);
}
__device__ __forceinline__ float leaky(float e) { return e > 0.0f ? e : LEAKY * e; }

// =====================================================================
// Fused triple GEMM: H{gat,gcn,lin} = h @ W{gat,gcn,lin}  via WMMA f32 16x16x4
// grid = (ceil(N/64), 3), block = 256 (8 waves).
// Each wave: 64 rows x 16 cols = 4 row-subtiles register-blocked so each
// B fragment load feeds 4 WMMAs (loads/WMMA: 1.5 instead of 3).
// A frag: lane l -> row, VGPR v -> K = kb + v + 2*(l>>4)
// B frag: lane l -> col=colbase+(l&15), VGPR v -> K = kb + v + 2*(l>>4)
// C/D   : lane l -> col=colbase+(l&15), VGPR v -> row = subbase + v + 8*(l>>4)
// Out-of-range rows are clamped on load (EXEC stays all-1s for WMMA) and
// masked on store.
// =====================================================================
__global__ void k_gemm3(const float* __restrict__ h,
                        const float* __restrict__ Wg,
                        const float* __restrict__ Wc,
                        const float* __restrict__ Wl,
                        float* __restrict__ Hg, float* __restrict__ Hc,
                        float* __restrict__ Hl, int n_nodes) {
    const int lane    = threadIdx.x & 31;
    const int wave    = threadIdx.x >> 5;
    const int rowbase = blockIdx.x * 64;
    const int mat     = blockIdx.y;
    const int colbase = wave * 16;

    const float* W  = (mat == 0) ? Wg : (mat == 1) ? Wc : Wl;
    float*       Ho = (mat == 0) ? Hg : (mat == 1) ? Hc : Hl;

    const int hh  = lane >> 4;
    const int col = colbase + (lane & 15);

    const float* arow[4];
#pragma unroll
    for (int s = 0; s < 4; ++s) {
        int r = rowbase + s * 16 + (lane & 15);
        if (r >= n_nodes) r = n_nodes - 1;  // clamp: load valid data, mask at store
        arow[s] = h + (size_t)r * D + 2 * hh;
    }
    const float* bcol = W + (size_t)(2 * hh) * D + col;

    v8f c[4] = {v8f{}, v8f{}, v8f{}, v8f{}};
#pragma unroll
    for (int kb = 0; kb < D; kb += 4) {
        v2f b;
        b.x = bcol[(size_t)kb * D];
        b.y = bcol[(size_t)(kb + 1) * D];
#pragma unroll
        for (int s = 0; s < 4; ++s) {
            v2f a;
            a.x  = arow[s][kb + 0];
            a.y  = arow[s][kb + 1];
            c[s] = __builtin_amdgcn_wmma_f32_16x16x4_f32(false, a, false, b,
                                                         (short)0, c[s], false, false);
        }
    }
#pragma unroll
    for (int s = 0; s < 4; ++s) {
#pragma unroll
        for (int v = 0; v < 8; ++v) {
            const int row = rowbase + s * 16 + v + 8 * hh;
            if (row < n_nodes) Ho[(size_t)row * D + col] = c[s][v];
        }
    }
}

// ---- per-node attention logits: as = Hg.a0 , ad = Hg.a1 ----
__global__ void k_scores(const float* __restrict__ Hg, const float* __restrict__ a0,
                         const float* __restrict__ a1, float* __restrict__ as_,
                         float* __restrict__ ad_, int n) {
    int i = blockIdx.x * blockDim.x + threadIdx.x;
    if (i >= n) return;
    const float4* hp = (const float4*)(Hg + (size_t)i * D);
    const float4* p0 = (const float4*)a0;
    const float4* p1 = (const float4*)a1;
    float s0 = 0.f, s1 = 0.f;
#pragma unroll 8
    for (int k = 0; k < D / 4; ++k) {
        float4 h4 = hp[k], x0 = p0[k], x1 = p1[k];
        s0 += h4.x * x0.x + h4.y * x0.y + h4.z * x0.z + h4.w * x0.w;
        s1 += h4.x * x1.x + h4.y * x1.y + h4.z * x1.z + h4.w * x1.w;
    }
    as_[i] = s0;
    ad_[i] = s1;
}

__global__ void k_fill_f32(float* p, float v, int n) {
    int i = blockIdx.x * blockDim.x + threadIdx.x;
    if (i < n) p[i] = v;
}
__global__ void k_fill_u32(unsigned* p, unsigned v, int n) {
    int i = blockIdx.x * blockDim.x + threadIdx.x;
    if (i < n) p[i] = v;
}

// ---- degree (deg pre-filled with 1.0 for self loop) ----
__global__ void k_deg(const int* __restrict__ dst, float* __restrict__ deg, int E) {
    int i = blockIdx.x * blockDim.x + threadIdx.x;
    if (i < E) atomicAdd(&deg[dst[i]], 1.0f);
}
__global__ void k_dinv(const float* __restrict__ deg, float* __restrict__ dinv, int n) {
    int i = blockIdx.x * blockDim.x + threadIdx.x;
    if (i < n) dinv[i] = rsqrtf(fmaxf(deg[i], 1.0f));
}

// ---- segment max of leaky(as[s]+ad[t]) over incoming edges (incl. self loops) ----
__global__ void k_edge_max(const int* __restrict__ src, const int* __restrict__ dst,
                           const float* __restrict__ as_, const float* __restrict__ ad_,
                           unsigned* __restrict__ m, int E, int n) {
    int i = blockIdx.x * blockDim.x + threadIdx.x;
    if (i >= E + n) return;
    int s, t;
    if (i < E) { s = src[i]; t = dst[i]; } else { s = t = i - E; }
    atomicMax(&m[t], f2ord(leaky(as_[s] + ad_[t])));
}

// ---- softmax denominator ----
__global__ void k_edge_den(const int* __restrict__ src, const int* __restrict__ dst,
                           const float* __restrict__ as_, const float* __restrict__ ad_,
                           const unsigned* __restrict__ m, float* __restrict__ den,
                           int E, int n) {
    int i = blockIdx.x * blockDim.x + threadIdx.x;
    if (i >= E + n) return;
    int s, t;
    if (i < E) { s = src[i]; t = dst[i]; } else { s = t = i - E; }
    float e = leaky(as_[s] + ad_[t]);
    atomicAdd(&den[t], expf(e - ord2f(m[t])));
}

// ---- acc init: biases + Hl + self-loop GAT + self-loop GCN (wave per node) ----
__global__ void k_acc_init(const float* __restrict__ Hg, const float* __restrict__ Hc,
                           const float* __restrict__ Hl, const float* __restrict__ as_,
                           const float* __restrict__ ad_, const unsigned* __restrict__ m,
                           const float* __restrict__ den, const float* __restrict__ dinv,
                           const float* __restrict__ bg, const float* __restrict__ bc,
                           const float* __restrict__ bl, float* __restrict__ acc, int n) {
    int node = blockIdx.x * (blockDim.x >> 5) + (threadIdx.x >> 5);
    int lane = threadIdx.x & 31;
    if (node >= n) return;
    float e     = leaky(as_[node] + ad_[node]);
    float alpha = expf(e - ord2f(m[node])) / den[node];
    float g     = dinv[node];
    g           = g * g;
    size_t off = (size_t)node * D + lane * 4;
    int    d0  = lane * 4;
    float4 hg = *(const float4*)(Hg + off);
    float4 hc = *(const float4*)(Hc + off);
    float4 hl = *(const float4*)(Hl + off);
    float4 o;
    o.x = hl.x + bg[d0 + 0] + bc[d0 + 0] + bl[d0 + 0] + alpha * hg.x + g * hc.x;
    o.y = hl.y + bg[d0 + 1] + bc[d0 + 1] + bl[d0 + 1] + alpha * hg.y + g * hc.y;
    o.z = hl.z + bg[d0 + 2] + bc[d0 + 2] + bl[d0 + 2] + alpha * hg.z + g * hc.z;
    o.w = hl.w + bg[d0 + 3] + bc[d0 + 3] + bl[d0 + 3] + alpha * hg.w + g * hc.w;
    *(float4*)(acc + off) = o;
}

// ---- fused GAT+GCN edge scatter (wave per edge, 4 dims per lane) ----
__global__ void k_edge_scatter(const int* __restrict__ src, const int* __restrict__ dst,
                               const float* __restrict__ as_, const float* __restrict__ ad_,
                               const unsigned* __restrict__ m, const float* __restrict__ den,
                               const float* __restrict__ dinv, const float* __restrict__ Hg,
                               const float* __restrict__ Hc, float* __restrict__ acc, int E) {
    int e    = blockIdx.x * (blockDim.x >> 5) + (threadIdx.x >> 5);
    int lane = threadIdx.x & 31;
    if (e >= E) return;
    int   s     = src[e], t = dst[e];
    float sc    = leaky(as_[s] + ad_[t]);
    float alpha = expf(sc - ord2f(m[t])) / den[t];
    float g     = dinv[s] * dinv[t];
    size_t so = (size_t)s * D + lane * 4;
    size_t to = (size_t)t * D + lane * 4;
    float4 hg = *(const float4*)(Hg + so);
    float4 hc = *(const float4*)(Hc + so);
    atomicAdd(acc + to + 0, alpha * hg.x + g * hc.x);
    atomicAdd(acc + to + 1, alpha * hg.y + g * hc.y);
    atomicAdd(acc + to + 2, alpha * hg.z + g * hc.z);
    atomicAdd(acc + to + 3, alpha * hg.w + g * hc.w);
}

__global__ void k_relu(float* __restrict__ p, int nelem) {
    int i = blockIdx.x * blockDim.x + threadIdx.x;
    if (i * 4 >= nelem) return;
    float4 v = *(float4*)(p + (size_t)i * 4);
    v.x = fmaxf(v.x, 0.f); v.y = fmaxf(v.y, 0.f);
    v.z = fmaxf(v.z, 0.f); v.w = fmaxf(v.w, 0.f);
    *(float4*)(p + (size_t)i * 4) = v;
}

// ---- graph mean pool (wave per node) ----
__global__ void k_pool(const float* __restrict__ h, const int* __restrict__ batch,
                       float* __restrict__ pooled, float* __restrict__ cnt, int n) {
    int node = blockIdx.x * (blockDim.x >> 5) + (threadIdx.x >> 5);
    int lane = threadIdx.x & 31;
    if (node >= n) return;
    int g = batch[node];
    if (lane == 0) atomicAdd(&cnt[g], 1.0f);
    float4 v = *(const float4*)(h + (size_t)node * D + lane * 4);
    float* pg = pooled + (size_t)g * D + lane * 4;
    atomicAdd(pg + 0, v.x);
    atomicAdd(pg + 1, v.y);
    atomicAdd(pg + 2, v.z);
    atomicAdd(pg + 3, v.w);
}

// ---- classifier head: out[g][o] = mean_pool . W4 + b4 ----
__global__ void k_out(const float* __restrict__ pooled, const float* __restrict__ cnt,
                      const float* __restrict__ W4, const float* __restrict__ b4,
                      float* __restrict__ out, int ngraphs) {
    int i = threadIdx.x;
    int g = i >> 1, o = i & 1;
    if (g >= ngraphs) return;
    float c = fmaxf(cnt[g], 1.0f);
    float s = 0.f;
    for (int d = 0; d < D; ++d) s += pooled[(size_t)g * D + d] * W4[d * 2 + o];
    out[g * 2 + o] = s / c + b4[o];
}

extern "C" void kernel_launch(void* const* d_in, const int* in_sizes, int n_in,
                              void* d_out, int out_size, void* d_ws, size_t ws_size,
                              hipStream_t stream) {
    const float* x    = (const float*)d_in[0];
    const float* gatW = (const float*)d_in[1];
    const float* gatA = (const float*)d_in[2];
    const float* gatB = (const float*)d_in[3];
    const float* gcnW = (const float*)d_in[4];
    const float* gcnB = (const float*)d_in[5];
    const float* linW = (const float*)d_in[6];
    const float* linB = (const float*)d_in[7];
    const float* W4   = (const float*)d_in[8];
    const float* b4   = (const float*)d_in[9];
    const int*   src  = (const int*)d_in[10];
    const int*   dst  = (const int*)d_in[11];
    const int*   bat  = (const int*)d_in[12];

    const int E  = in_sizes[10];
    const int N  = in_sizes[12];
    const int NG = out_size / 2;

    // workspace layout (floats)
    const size_t NE = (size_t)N * D;
    float*    ws     = (float*)d_ws;
    float*    Hg     = ws;
    float*    Hc     = Hg + NE;
    float*    Hl     = Hc + NE;
    float*    hA     = Hl + NE;
    float*    hB     = hA + NE;
    float*    deg    = hB + NE;
    float*    dinv   = deg + N;
    float*    as_    = dinv + N;
    float*    ad_    = as_ + N;
    unsigned* m      = (unsigned*)(ad_ + N);
    float*    den    = (float*)(m + N);
    float*    pooled = den + N;
    float*    cnt    = pooled + (size_t)NG * D;

    auto cdiv = [](int a, int b) { return (a + b - 1) / b; };

    // degree / dinv (layer invariant; self loop contributes the initial 1.0)
    k_fill_f32<<<cdiv(N, 256), 256, 0, stream>>>(deg, 1.0f, N);
    k_deg<<<cdiv(E, 256), 256, 0, stream>>>(dst, deg, E);
    k_dinv<<<cdiv(N, 256), 256, 0, stream>>>(deg, dinv, N);

    for (int i = 0; i < 3; ++i) {
        const float* hsrc = (i == 0) ? x : (i == 1) ? hA : hB;
        float*       acc  = (i == 0) ? hA : (i == 1) ? hB : hA;

        k_gemm3<<<dim3(cdiv(N, 64), 3), 256, 0, stream>>>(
            hsrc, gatW + (size_t)i * D * D, gcnW + (size_t)i * D * D,
            linW + (size_t)i * D * D, Hg, Hc, Hl, N);
        k_scores<<<cdiv(N, 256), 256, 0, stream>>>(
            Hg, gatA + (size_t)i * 2 * D, gatA + (size_t)i * 2 * D + D, as_, ad_, N);
        k_fill_u32<<<cdiv(N, 256), 256, 0, stream>>>(m, 0u, N);
        k_fill_f32<<<cdiv(N, 256), 256, 0, stream>>>(den, 0.f, N);
        k_edge_max<<<cdiv(E + N, 256), 256, 0, stream>>>(src, dst, as_, ad_, m, E, N);
        k_edge_den<<<cdiv(E + N, 256), 256, 0, stream>>>(src, dst, as_, ad_, m, den, E, N);
        k_acc_init<<<cdiv(N, 8), 256, 0, stream>>>(
            Hg, Hc, Hl, as_, ad_, m, den, dinv,
            gatB + (size_t)i * D, gcnB + (size_t)i * D, linB + (size_t)i * D, acc, N);
        k_edge_scatter<<<cdiv(E, 8), 256, 0, stream>>>(
            src, dst, as_, ad_, m, den, dinv, Hg, Hc, acc, E);
        if (i < 2) k_relu<<<cdiv(N * D / 4, 256), 256, 0, stream>>>(acc, N * D);
    }

    k_fill_f32<<<cdiv(NG * D, 256), 256, 0, stream>>>(pooled, 0.f, NG * D);
    k_fill_f32<<<1, 256, 0, stream>>>(cnt, 0.f, NG);
    k_pool<<<cdiv(N, 8), 256, 0, stream>>>(hA, bat, pooled, cnt, N);
    k_out<<<1, 128, 0, stream>>>(pooled, cnt, W4, b4, (float*)d_out, NG);
}